// LogSlaterDeterminant_76312978915891
// MI455X (gfx1250) — compile-verified
//
#include <hip/hip_runtime.h>
#include <hip/hip_bf16.h>
#include <math.h>

typedef __attribute__((ext_vector_type(2))) float v2f;
typedef __attribute__((ext_vector_type(8))) float v8f;

#define NORB 256
#define NF   64
#define LDA  68   // padded LDS row stride (floats) to dodge bank conflicts

__global__ __launch_bounds__(256)
void LogSlaterDeterminant_76312978915891_kernel(const int* __restrict__ nmask,
                                                const float* __restrict__ kernelf, // interleaved complex [NF][NORB]
                                                float* __restrict__ out)           // interleaved complex [B]
{
    __shared__ float sAr[NF * LDA];
    __shared__ float sAi[NF * LDA];
    __shared__ int   sOcc[NF];
    __shared__ int   sScan[256];
    __shared__ float sRedV[64];
    __shared__ int   sRedI[64];
    __shared__ float sInv[2];

    const int tid = threadIdx.x;
    const int b   = blockIdx.x;
    const int lid = tid & 31;
    const int wid = tid >> 5;

    // ---- 1. occupied-orbital indices via inclusive prefix scan ----
    int flag = (nmask[b * NORB + tid] != 0) ? 1 : 0;
    sScan[tid] = flag;
    __syncthreads();
    for (int off = 1; off < 256; off <<= 1) {
        int v = (tid >= off) ? sScan[tid - off] : 0;
        __syncthreads();
        sScan[tid] += v;
        __syncthreads();
    }
    if (flag) {
        int pos = sScan[tid] - 1;
        if (pos < NF) sOcc[pos] = tid;
    }
    __syncthreads();

    // ---- 2. gather phi[i][k] = kernel[i, occ[k]] into LDS (split re/im planes) ----
    for (int idx = tid; idx < NF * NF; idx += 256) {
        int i = idx >> 6, k = idx & 63;
        int base = 2 * (i * NORB + sOcc[k]);
        sAr[i * LDA + k] = kernelf[base];
        sAi[i * LDA + k] = kernelf[base + 1];
    }
    __syncthreads();

    float logabs = 0.0f, sr = 1.0f, si = 0.0f;  // det accumulators (thread 0)

    for (int pb = 0; pb < NF; pb += 16) {
        const int pe = pb + 16;
        // ---- 3. panel factorization with partial pivoting ----
        for (int j = pb; j < pe; ++j) {
            // pivot argmax |A[r][j]|^2 over r in [j,64)
            if (tid < 64) {
                float v = -1.0f;
                if (tid >= j) {
                    float xr = sAr[tid * LDA + j], xi = sAi[tid * LDA + j];
                    v = xr * xr + xi * xi;
                }
                sRedV[tid] = v; sRedI[tid] = tid;
            }
            __syncthreads();
            for (int s = 32; s >= 1; s >>= 1) {
                if (tid < s) {
                    if (sRedV[tid + s] > sRedV[tid]) {
                        sRedV[tid] = sRedV[tid + s];
                        sRedI[tid] = sRedI[tid + s];
                    }
                }
                __syncthreads();
            }
            int piv = sRedI[0];
            if (piv != j && tid < 64) {  // full-row swap
                float tr = sAr[j * LDA + tid];
                sAr[j * LDA + tid] = sAr[piv * LDA + tid]; sAr[piv * LDA + tid] = tr;
                float ti = sAi[j * LDA + tid];
                sAi[j * LDA + tid] = sAi[piv * LDA + tid]; sAi[piv * LDA + tid] = ti;
            }
            __syncthreads();
            if (tid == 0) {
                float ar_ = sAr[j * LDA + j], ai_ = sAi[j * LDA + j];
                float m2 = ar_ * ar_ + ai_ * ai_;
                logabs += 0.5f * logf(m2);
                float ia = rsqrtf(m2);
                float ur = ar_ * ia, ui = ai_ * ia;            // unit phasor of u_jj
                float nsr = sr * ur - si * ui;
                float nsi = sr * ui + si * ur;
                sr = nsr; si = nsi;
                if (piv != j) { sr = -sr; si = -si; }          // swap parity
                sInv[0] = ar_ / m2; sInv[1] = -ai_ / m2;       // 1/u_jj = conj/|u|^2
            }
            __syncthreads();
            float ivr = sInv[0], ivi = sInv[1];
            if (tid < 64 && tid > j) {                          // scale L column
                float xr = sAr[tid * LDA + j], xi = sAi[tid * LDA + j];
                sAr[tid * LDA + j] = xr * ivr - xi * ivi;
                sAi[tid * LDA + j] = xr * ivi + xi * ivr;
            }
            __syncthreads();
            // rank-1 update restricted to panel columns, all rows below j
            int pcols = pe - 1 - j;
            int nrows = 63 - j;
            int tot = pcols * nrows;
            for (int idx = tid; idx < tot; idx += 256) {
                int r = j + 1 + idx / pcols;
                int c = j + 1 + idx % pcols;
                float lr = sAr[r * LDA + j], li = sAi[r * LDA + j];
                float ur = sAr[j * LDA + c], ui = sAi[j * LDA + c];
                sAr[r * LDA + c] -= lr * ur - li * ui;
                sAi[r * LDA + c] -= lr * ui + li * ur;
            }
            __syncthreads();
        }
        int ntr = NF - pe;
        if (ntr > 0) {
            // ---- 4. U12 = L11^{-1} A12 (sequential rank-1s inside panel rows) ----
            for (int j = pb; j < pe - 1; ++j) {
                int nrows = pe - 1 - j;
                int tot = nrows * ntr;
                for (int idx = tid; idx < tot; idx += 256) {
                    int r = j + 1 + idx / ntr;
                    int c = pe + idx % ntr;
                    float lr = sAr[r * LDA + j], li = sAi[r * LDA + j];
                    float ur = sAr[j * LDA + c], ui = sAi[j * LDA + c];
                    sAr[r * LDA + c] -= lr * ur - li * ui;
                    sAi[r * LDA + c] -= lr * ui + li * ur;
                }
                __syncthreads();
            }
            // ---- 5. A22 -= L21 * U12 : complex GEMM via V_WMMA_F32_16X16X4_F32 ----
            int tdim = ntr >> 4;
            int ntiles = tdim * tdim;
            for (int t = wid; t < ntiles; t += 8) {            // wave-uniform tile loop
                int rt = pe + (t / tdim) * 16;
                int ct = pe + (t % tdim) * 16;
                int cm = (lid >> 4) * 8;                       // C/D: row = cm+r, col = cn
                int cn = lid & 15;
                v8f cr, ci;
#pragma unroll
                for (int r = 0; r < 8; ++r) {
                    cr[r] = sAr[(rt + cm + r) * LDA + ct + cn];
                    ci[r] = sAi[(rt + cm + r) * LDA + ct + cn];
                }
                int am = lid & 15;                             // A: row = am, k = ak+v
                int ak = (lid >> 4) * 2;
#pragma unroll
                for (int k0 = 0; k0 < 16; k0 += 4) {
                    v2f arf, aif, brf, bif;
#pragma unroll
                    for (int v = 0; v < 2; ++v) {
                        arf[v] = sAr[(rt + am) * LDA + pb + k0 + ak + v];
                        aif[v] = sAi[(rt + am) * LDA + pb + k0 + ak + v];
                        brf[v] = sAr[(pb + k0 + ak + v) * LDA + ct + cn];
                        bif[v] = sAi[(pb + k0 + ak + v) * LDA + ct + cn];
                    }
                    v2f narf = -arf;
                    v2f naif = -aif;
                    // Cr += (-Ar)*Br + Ai*Bi ; Ci += (-Ar)*Bi + (-Ai)*Br
                    cr = __builtin_amdgcn_wmma_f32_16x16x4_f32(false, narf, false, brf, (short)0, cr, false, false);
                    cr = __builtin_amdgcn_wmma_f32_16x16x4_f32(false, aif,  false, bif, (short)0, cr, false, false);
                    ci = __builtin_amdgcn_wmma_f32_16x16x4_f32(false, narf, false, bif, (short)0, ci, false, false);
                    ci = __builtin_amdgcn_wmma_f32_16x16x4_f32(false, naif, false, brf, (short)0, ci, false, false);
                }
#pragma unroll
                for (int r = 0; r < 8; ++r) {
                    sAr[(rt + cm + r) * LDA + ct + cn] = cr[r];
                    sAi[(rt + cm + r) * LDA + ct + cn] = ci[r];
                }
            }
            __syncthreads();
        }
    }

    if (tid == 0) {
        out[2 * b]     = logabs;
        out[2 * b + 1] = atan2f(si, sr);
    }
}

extern "C" void kernel_launch(void* const* d_in, const int* in_sizes, int n_in,
                              void* d_out, int out_size, void* d_ws, size_t ws_size,
                              hipStream_t stream) {
    (void)in_sizes; (void)n_in; (void)d_ws; (void)ws_size; (void)out_size;
    const int*   nmask   = (const int*)d_in[0];
    const float* kernelf = (const float*)d_in[1];  // complex64 interleaved
    float*       out     = (float*)d_out;          // complex64 interleaved [B]
    const int B = 8192;
    LogSlaterDeterminant_76312978915891_kernel<<<B, 256, 0, stream>>>(nmask, kernelf, out);
}